// ClimateConditionedGAT_28518582846049
// MI455X (gfx1250) — compile-verified
//
#include <hip/hip_runtime.h>
#include <hip/hip_bf16.h>
#include <math.h>

typedef __attribute__((ext_vector_type(2))) float v2f;
typedef __attribute__((ext_vector_type(8))) float v8f;

#define HEADS 4
#define F_OUT 64
#define D_IN 128
#define NH_COLS 256   // HEADS * F_OUT
#define NEG_SLOPE 0.2f

// ---------------------------------------------------------------------------
// Kernel 1: h = x @ W_gat  via V_WMMA_F32_16X16X4_F32 (full fp32 precision).
// One block = 128 threads = 4 waves; each wave computes a 16x64 output strip.
// A-fragment (16x4 tile of x) is loaded once per k-step and reused for 4 WMMAs.
// ---------------------------------------------------------------------------
__global__ __launch_bounds__(128) void gat_gemm_wmma(
    const float* __restrict__ X,   // [N,128]
    const float* __restrict__ W,   // [128,256]
    float* __restrict__ H)         // [N,256]
{
    const int mtile   = blockIdx.x;            // 16-row tile index (N % 16 == 0)
    const int wave    = threadIdx.x >> 5;      // 0..3 -> 64-col strip
    const int lane    = threadIdx.x & 31;
    const int halfsel = lane >> 4;             // 0: K={0,1}, 1: K={2,3}
    const int l16     = lane & 15;
    const int colbase = wave * 64;

    const float* xrow = X + (size_t)(mtile * 16 + l16) * D_IN;

    v8f acc0 = {}; v8f acc1 = {}; v8f acc2 = {}; v8f acc3 = {};

    #pragma unroll 4
    for (int s = 0; s < 32; ++s) {
        const int k0 = s * 4 + halfsel * 2;
        // A fragment: 16x4 f32 tile, lane holds (row=l16, K=k0,k0+1) -> b64 load
        v2f a;
        a.x = xrow[k0];
        a.y = xrow[k0 + 1];
        // B fragments: column-major layout, lane holds (col, K=k0,k0+1)
        const float* w0 = W + (size_t)k0 * NH_COLS;
        const float* w1 = W + (size_t)(k0 + 1) * NH_COLS;
        const int c0 = colbase + l16;
        v2f b0; b0.x = w0[c0     ]; b0.y = w1[c0     ];
        v2f b1; b1.x = w0[c0 + 16]; b1.y = w1[c0 + 16];
        v2f b2; b2.x = w0[c0 + 32]; b2.y = w1[c0 + 32];
        v2f b3; b3.x = w0[c0 + 48]; b3.y = w1[c0 + 48];

        acc0 = __builtin_amdgcn_wmma_f32_16x16x4_f32(false, a, false, b0, (short)0, acc0, false, false);
        acc1 = __builtin_amdgcn_wmma_f32_16x16x4_f32(false, a, false, b1, (short)0, acc1, false, false);
        acc2 = __builtin_amdgcn_wmma_f32_16x16x4_f32(false, a, false, b2, (short)0, acc2, false, false);
        acc3 = __builtin_amdgcn_wmma_f32_16x16x4_f32(false, a, false, b3, (short)0, acc3, false, false);
    }

    // C/D layout: VGPR r holds M = r + 8*halfsel, N = l16 (per 16-col sub-tile)
    #pragma unroll
    for (int r = 0; r < 8; ++r) {
        const int m = mtile * 16 + r + halfsel * 8;
        float* hrow = H + (size_t)m * NH_COLS + colbase + l16;
        hrow[0]  = acc0[r];
        hrow[16] = acc1[r];
        hrow[32] = acc2[r];
        hrow[48] = acc3[r];
    }
}

// ---------------------------------------------------------------------------
// Kernel 0: init amax = -inf, denom = 0, agg = 0
// ---------------------------------------------------------------------------
__global__ void gat_init(float* __restrict__ amax, float* __restrict__ denom,
                         float* __restrict__ agg, int nh, int naggr)
{
    const int i = blockIdx.x * blockDim.x + threadIdx.x;
    if (i < nh) {
        amax[i]  = __int_as_float(0xFF800000); // -inf
        denom[i] = 0.0f;
    }
    if (i < naggr) agg[i] = 0.0f;
}

// ---------------------------------------------------------------------------
// Kernel 2: a_src[n,h] = <h[n,h,:], att_src[h,:]>, same for a_dst
// ---------------------------------------------------------------------------
__global__ void gat_attn_coef(const float* __restrict__ H,
                              const float* __restrict__ att_src,
                              const float* __restrict__ att_dst,
                              float* __restrict__ a_src,
                              float* __restrict__ a_dst, int nh)
{
    const int tid = blockIdx.x * blockDim.x + threadIdx.x;
    if (tid >= nh) return;
    const int n = tid >> 2, hd = tid & 3;
    const float4* hp = (const float4*)(H + (size_t)n * NH_COLS + hd * F_OUT);
    const float4* as = (const float4*)(att_src + hd * F_OUT);
    const float4* ad = (const float4*)(att_dst + hd * F_OUT);
    float s1 = 0.0f, s2 = 0.0f;
    #pragma unroll
    for (int i = 0; i < 16; ++i) {
        const float4 hv = hp[i], v1 = as[i], v2 = ad[i];
        s1 += hv.x * v1.x + hv.y * v1.y + hv.z * v1.z + hv.w * v1.w;
        s2 += hv.x * v2.x + hv.y * v2.y + hv.z * v2.z + hv.w * v2.w;
    }
    a_src[tid] = s1;
    a_dst[tid] = s2;
}

// Float atomic max via signed-max / unsigned-min bit trick -> native L2 atomics
__device__ __forceinline__ void atomicMaxFloat(float* addr, float val)
{
    if (val >= 0.0f) atomicMax((int*)addr, __float_as_int(val));
    else             atomicMin((unsigned int*)addr, __float_as_uint(val));
}

__device__ __forceinline__ void edge_endpoints(const int* __restrict__ ei,
                                               int e, int E, int& src, int& dst)
{
    if (e < E) { src = ei[e]; dst = ei[E + e]; }
    else       { src = dst = e - E; }          // self loop
}

__device__ __forceinline__ float leaky(float a)
{
    return a > 0.0f ? a : NEG_SLOPE * a;
}

// ---------------------------------------------------------------------------
// Kernel 3: segment max of leaky_relu(a_src[src]+a_dst[dst]) over dst
// ---------------------------------------------------------------------------
__global__ void gat_edge_max(const int* __restrict__ ei, int E, int Etot,
                             const float* __restrict__ a_src,
                             const float* __restrict__ a_dst,
                             float* __restrict__ amax)
{
    const int tid = blockIdx.x * blockDim.x + threadIdx.x;
    if (tid >= Etot * HEADS) return;
    const int e = tid >> 2, hd = tid & 3;
    int src, dst; edge_endpoints(ei, e, E, src, dst);
    const float al = leaky(a_src[src * HEADS + hd] + a_dst[dst * HEADS + hd]);
    atomicMaxFloat(&amax[dst * HEADS + hd], al);
}

// ---------------------------------------------------------------------------
// Kernel 4: denom[dst,h] += exp(alpha - amax[dst,h])
// ---------------------------------------------------------------------------
__global__ void gat_edge_denom(const int* __restrict__ ei, int E, int Etot,
                               const float* __restrict__ a_src,
                               const float* __restrict__ a_dst,
                               const float* __restrict__ amax,
                               float* __restrict__ denom)
{
    const int tid = blockIdx.x * blockDim.x + threadIdx.x;
    if (tid >= Etot * HEADS) return;
    const int e = tid >> 2, hd = tid & 3;
    int src, dst; edge_endpoints(ei, e, E, src, dst);
    const float al = leaky(a_src[src * HEADS + hd] + a_dst[dst * HEADS + hd]);
    atomicAdd(&denom[dst * HEADS + hd], __expf(al - amax[dst * HEADS + hd]));
}

// ---------------------------------------------------------------------------
// Kernel 5: agg[dst] += w * h[src]; 64 threads per edge, float4 per thread.
// 1.7 GB of read + L2-resident fp32 atomics -> the dominant stage.
// ---------------------------------------------------------------------------
__global__ void gat_edge_agg(const int* __restrict__ ei, int E, int Etot,
                             const float* __restrict__ a_src,
                             const float* __restrict__ a_dst,
                             const float* __restrict__ amax,
                             const float* __restrict__ denom,
                             const float* __restrict__ H,
                             float* __restrict__ agg)
{
    const int gid = blockIdx.x * blockDim.x + threadIdx.x;
    const int e = gid >> 6;
    if (e >= Etot) return;
    const int t  = gid & 63;
    const int hd = t >> 4;           // head
    const int f  = (t & 15) * 4;     // float4 offset within head
    int src, dst; edge_endpoints(ei, e, E, src, dst);
    const int sh = src * HEADS + hd, dh = dst * HEADS + hd;
    const float al = leaky(a_src[sh] + a_dst[dh]);
    const float w  = __expf(al - amax[dh]) / (denom[dh] + 1e-16f);
    const float4 hv = *(const float4*)(H + (size_t)src * NH_COLS + hd * F_OUT + f);
    float* o = agg + (size_t)dst * NH_COLS + hd * F_OUT + f;
    atomicAdd(o + 0, w * hv.x);
    atomicAdd(o + 1, w * hv.y);
    atomicAdd(o + 2, w * hv.z);
    atomicAdd(o + 3, w * hv.w);
}

// ---------------------------------------------------------------------------
// Kernel 6: out = (1-g) * (head_mean(agg) + bias_gat) + g * (onehot@W_prior + b_prior)
// ---------------------------------------------------------------------------
__global__ void gat_finalize(const float* __restrict__ agg,
                             const float* __restrict__ bias_gat,
                             const float* __restrict__ onehot,   // [N,32]
                             const float* __restrict__ W_prior,  // [32,64]
                             const float* __restrict__ b_prior,
                             const float* __restrict__ gate,
                             float* __restrict__ out, int nf)
{
    const int tid = blockIdx.x * blockDim.x + threadIdx.x;
    if (tid >= nf) return;
    const int n = tid >> 6, f = tid & 63;
    const float g = gate[0];
    const float* ar = agg + (size_t)n * NH_COLS + f;
    const float m = 0.25f * (ar[0] + ar[64] + ar[128] + ar[192]) + bias_gat[f];
    float p = b_prior[f];
    const float* oh = onehot + (size_t)n * 32;
    #pragma unroll
    for (int c = 0; c < 32; ++c) p += oh[c] * W_prior[c * 64 + f];
    out[tid] = (1.0f - g) * m + g * p;
}

// ---------------------------------------------------------------------------
extern "C" void kernel_launch(void* const* d_in, const int* in_sizes, int n_in,
                              void* d_out, int out_size, void* d_ws, size_t ws_size,
                              hipStream_t stream)
{
    const float* x        = (const float*)d_in[0];
    const int*   ei       = (const int*)  d_in[1];
    const float* onehot   = (const float*)d_in[2];
    const float* W_gat    = (const float*)d_in[3];
    const float* att_src  = (const float*)d_in[4];
    const float* att_dst  = (const float*)d_in[5];
    const float* bias_gat = (const float*)d_in[6];
    const float* W_prior  = (const float*)d_in[7];
    const float* b_prior  = (const float*)d_in[8];
    const float* gate     = (const float*)d_in[9];
    float* out = (float*)d_out;

    const int N    = in_sizes[0] / D_IN;   // 50000
    const int E    = in_sizes[1] / 2;      // 800000
    const int Etot = E + N;                // + self loops
    const int NHs  = N * HEADS;

    // Workspace layout (floats)
    float* h      = (float*)d_ws;               // N*256
    float* a_src  = h      + (size_t)N * NH_COLS;
    float* a_dst  = a_src  + NHs;
    float* amax   = a_dst  + NHs;
    float* denom  = amax   + NHs;
    float* agg    = denom  + NHs;               // N*256

    const int naggr = N * NH_COLS;

    gat_init<<<(naggr + 255) / 256, 256, 0, stream>>>(amax, denom, agg, NHs, naggr);

    gat_gemm_wmma<<<N / 16, 128, 0, stream>>>(x, W_gat, h);

    gat_attn_coef<<<(NHs + 255) / 256, 256, 0, stream>>>(h, att_src, att_dst,
                                                         a_src, a_dst, NHs);

    const int eh = Etot * HEADS;
    gat_edge_max<<<(eh + 255) / 256, 256, 0, stream>>>(ei, E, Etot, a_src, a_dst, amax);
    gat_edge_denom<<<(eh + 255) / 256, 256, 0, stream>>>(ei, E, Etot, a_src, a_dst,
                                                         amax, denom);

    const long long aggthreads = (long long)Etot * 64;
    gat_edge_agg<<<(unsigned)((aggthreads + 255) / 256), 256, 0, stream>>>(
        ei, E, Etot, a_src, a_dst, amax, denom, h, agg);

    const int nf = N * F_OUT;
    gat_finalize<<<(nf + 255) / 256, 256, 0, stream>>>(agg, bias_gat, onehot,
                                                       W_prior, b_prior, gate,
                                                       out, nf);
}